// GCN_49632642073097
// MI455X (gfx1250) — compile-verified
//
#include <hip/hip_runtime.h>
#include <math.h>

typedef __attribute__((ext_vector_type(2))) float v2f;
typedef __attribute__((ext_vector_type(8))) float v8f;

#define BN_EPS 1e-5f

// ---------------- utility kernels ----------------

__global__ void zero_f32(float* __restrict__ p, size_t n) {
  size_t i = (size_t)blockIdx.x * blockDim.x + threadIdx.x;
  size_t stride = (size_t)gridDim.x * blockDim.x;
  for (; i < n; i += stride) p[i] = 0.0f;
}

__global__ void degree_kernel(const int* __restrict__ src, const int* __restrict__ dst,
                              float* __restrict__ dout, float* __restrict__ din, int ne) {
  int i = blockIdx.x * blockDim.x + threadIdx.x;
  int stride = gridDim.x * blockDim.x;
  for (; i < ne; i += stride) {
    atomicAdd(&dout[src[i]], 1.0f);
    atomicAdd(&din[dst[i]], 1.0f);
  }
}

__global__ void norm_kernel(const float* __restrict__ dout, const float* __restrict__ din,
                            float* __restrict__ ns, float* __restrict__ nd, int n) {
  int i = blockIdx.x * blockDim.x + threadIdx.x;
  int stride = gridDim.x * blockDim.x;
  for (; i < n; i += stride) {
    ns[i] = rsqrtf(fmaxf(dout[i], 1.0f));
    nd[i] = rsqrtf(fmaxf(din[i], 1.0f));
  }
}

// hs[row,col] = x[row,col] * norm_s[row]   (128 cols, blockDim.x == 128)
__global__ void scale_kernel(const float* __restrict__ x, const float* __restrict__ ns,
                             float* __restrict__ out, int n) {
  int col = threadIdx.x;
  for (int row = blockIdx.x; row < n; row += gridDim.x) {
    size_t idx = (size_t)row * 128 + col;
    out[idx] = x[idx] * ns[row];
  }
}

// one wave per edge (grid-stride): float4 gather from L2-resident hs, 4 f32 atomics to agg
__global__ void spmm_kernel(const float* __restrict__ hs, const int* __restrict__ src,
                            const int* __restrict__ dst, float* __restrict__ agg, int ne) {
  int lane = threadIdx.x & 31;
  int w = (blockIdx.x * blockDim.x + threadIdx.x) >> 5;
  int nw = (gridDim.x * blockDim.x) >> 5;
  for (int e = w; e < ne; e += nw) {
    int s = src[e];
    int d = dst[e];
    const float4 v = *reinterpret_cast<const float4*>(hs + (size_t)s * 128 + lane * 4);
    float* p = agg + (size_t)d * 128 + lane * 4;
    atomicAdd(p + 0, v.x);
    atomicAdd(p + 1, v.y);
    atomicAdd(p + 2, v.z);
    atomicAdd(p + 3, v.w);
  }
}

// ---------------- WMMA fp32 GEMM: out[n,FOUT] = (A * nd[row]) @ W + bias ----------------
// One wave computes a 16-row x FOUT strip. K=128 swept in steps of 4 with
// v_wmma_f32_16x16x4_f32. A layout (16x4 fp32): lanes 0-15 carry K0/K1 of row
// M0+lane, lanes 16-31 carry K2/K3 of row M0+(lane-16). B (4x16) mirrors that split.
template <int FOUT, bool RELU>
__global__ __launch_bounds__(128) void gemm_wmma(const float* __restrict__ A,
                                                 const float* __restrict__ nd,
                                                 const float* __restrict__ W,
                                                 const float* __restrict__ bias,
                                                 float* __restrict__ out, int n) {
  __shared__ float sW[128 * FOUT];
  __shared__ float sB[FOUT];
  for (int i = threadIdx.x; i < 128 * FOUT; i += 128) sW[i] = W[i];
  if (threadIdx.x < FOUT) sB[threadIdx.x] = bias[threadIdx.x];
  __syncthreads();

  const int lane = threadIdx.x & 31;
  const int wave = threadIdx.x >> 5;
  const int tile = blockIdx.x * 4 + wave;
  const int ntiles = n >> 4;
  if (tile >= ntiles) return;  // whole wave exits together: EXEC all-1s for WMMA

  const int M0 = tile * 16;
  const int r = lane & 15;
  const int half = lane >> 4;   // 0: K0/K1, 1: K2/K3
  const int koff = half * 2;
  const int row = M0 + r;
  const float ndv = nd[row];
  const float2* A2 = reinterpret_cast<const float2*>(A + (size_t)row * 128);

  constexpr int NT = FOUT / 16;
  v8f acc[NT] = {};

#pragma unroll 4
  for (int k = 0; k < 128; k += 4) {
    float2 av = A2[(k + koff) >> 1];
    v2f a;
    a.x = av.x * ndv;
    a.y = av.y * ndv;
#pragma unroll
    for (int t = 0; t < NT; ++t) {
      int col = t * 16 + r;
      v2f b;
      b.x = sW[(k + koff) * FOUT + col];
      b.y = sW[(k + koff + 1) * FOUT + col];
      acc[t] = __builtin_amdgcn_wmma_f32_16x16x4_f32(
          /*neg_a=*/false, a, /*neg_b=*/false, b,
          /*c_mod=*/(short)0, acc[t], /*reuse_a=*/false, /*reuse_b=*/false);
    }
  }

  // C/D layout: VGPR i -> row M0+i (lanes 0-15) / M0+8+i (lanes 16-31), col = N0+(lane&15)
#pragma unroll
  for (int t = 0; t < NT; ++t) {
#pragma unroll
    for (int i = 0; i < 8; ++i) {
      int orow = M0 + i + half * 8;
      int col = t * 16 + r;
      float v = acc[t][i] + sB[col];
      if (RELU) v = fmaxf(v, 0.0f);
      out[(size_t)orow * FOUT + col] = v;
    }
  }
}

// ---------------- BatchNorm ----------------

// stats[0..127] = column sums, stats[128..255] = column sum-of-squares
__global__ void bn_stats(const float* __restrict__ h, float* __restrict__ stats, int n) {
  int col = threadIdx.x;  // blockDim.x == 128
  float s = 0.0f, q = 0.0f;
  for (int row = blockIdx.x; row < n; row += gridDim.x) {
    float v = h[(size_t)row * 128 + col];
    s += v;
    q += v * v;
  }
  atomicAdd(&stats[col], s);
  atomicAdd(&stats[128 + col], q);
}

// hs[row,col] = BN(h)[row,col] * norm_s[row]  (fused BN-normalize + next layer's src scale)
__global__ void bn_scale(const float* __restrict__ h, const float* __restrict__ stats,
                         const float* __restrict__ gamma, const float* __restrict__ beta,
                         const float* __restrict__ ns, float* __restrict__ out, int n) {
  int col = threadIdx.x;  // blockDim.x == 128
  float inv_n = 1.0f / (float)n;
  float mu = stats[col] * inv_n;
  float var = stats[128 + col] * inv_n - mu * mu;
  float g = gamma[col] * rsqrtf(var + BN_EPS);
  float b = beta[col];
  for (int row = blockIdx.x; row < n; row += gridDim.x) {
    size_t idx = (size_t)row * 128 + col;
    out[idx] = ((h[idx] - mu) * g + b) * ns[row];
  }
}

// ---------------- log-softmax over 64 cols, one wave per node ----------------
__global__ void logsoftmax_kernel(const float* __restrict__ h, float* __restrict__ out, int n) {
  int lane = threadIdx.x & 31;
  int w = (blockIdx.x * blockDim.x + threadIdx.x) >> 5;
  int nw = (gridDim.x * blockDim.x) >> 5;
  for (int node = w; node < n; node += nw) {
    const float2 v = *reinterpret_cast<const float2*>(h + (size_t)node * 64 + lane * 2);
    float m = fmaxf(v.x, v.y);
    for (int off = 16; off > 0; off >>= 1) m = fmaxf(m, __shfl_xor(m, off, 32));
    float s = expf(v.x - m) + expf(v.y - m);
    for (int off = 16; off > 0; off >>= 1) s += __shfl_xor(s, off, 32);
    float lse = m + logf(s);
    float2 o;
    o.x = v.x - lse;
    o.y = v.y - lse;
    *reinterpret_cast<float2*>(out + (size_t)node * 64 + lane * 2) = o;
  }
}

// ---------------- launch ----------------

extern "C" void kernel_launch(void* const* d_in, const int* in_sizes, int n_in,
                              void* d_out, int out_size, void* d_ws, size_t ws_size,
                              hipStream_t stream) {
  const float* x  = (const float*)d_in[0];
  const int* src  = (const int*)d_in[1];
  const int* dst  = (const int*)d_in[2];
  const float* W1 = (const float*)d_in[3];
  const float* b1 = (const float*)d_in[4];
  const float* W2 = (const float*)d_in[5];
  const float* b2 = (const float*)d_in[6];
  const float* W3 = (const float*)d_in[7];
  const float* b3 = (const float*)d_in[8];
  const float* g1 = (const float*)d_in[9];
  const float* be1 = (const float*)d_in[10];
  const float* g2 = (const float*)d_in[11];
  const float* be2 = (const float*)d_in[12];
  const int n  = in_sizes[0] / 128;
  const int ne = in_sizes[1];
  float* out = (float*)d_out;

  // workspace layout (floats)
  float* ws = (float*)d_ws;
  size_t off = 0;
  float* deg_out = ws + off; off += n;
  float* deg_in  = ws + off; off += n;
  float* norm_s  = ws + off; off += n;
  float* norm_d  = ws + off; off += n;
  float* stats   = ws + off; off += 256;
  float* h       = ws + off; off += (size_t)n * 128;
  float* hs      = ws + off; off += (size_t)n * 128;
  float* agg     = ws + off; off += (size_t)n * 128;

  const int ntiles = n / 16;
  const int gblocks = (ntiles + 3) / 4;

  // degrees + symmetric norms
  zero_f32<<<1024, 256, 0, stream>>>(deg_out, (size_t)2 * n);
  degree_kernel<<<2048, 256, 0, stream>>>(src, dst, deg_out, deg_in, ne);
  norm_kernel<<<1024, 256, 0, stream>>>(deg_out, deg_in, norm_s, norm_d, n);

  // ---- layer 1: conv(x) -> relu -> bn -> (scale for next layer) ----
  scale_kernel<<<4096, 128, 0, stream>>>(x, norm_s, hs, n);
  zero_f32<<<4096, 256, 0, stream>>>(agg, (size_t)n * 128);
  spmm_kernel<<<2048, 256, 0, stream>>>(hs, src, dst, agg, ne);
  gemm_wmma<128, true><<<gblocks, 128, 0, stream>>>(agg, norm_d, W1, b1, h, n);
  zero_f32<<<1, 256, 0, stream>>>(stats, 256);
  bn_stats<<<1024, 128, 0, stream>>>(h, stats, n);
  bn_scale<<<4096, 128, 0, stream>>>(h, stats, g1, be1, norm_s, hs, n);

  // ---- layer 2 ----
  zero_f32<<<4096, 256, 0, stream>>>(agg, (size_t)n * 128);
  spmm_kernel<<<2048, 256, 0, stream>>>(hs, src, dst, agg, ne);
  gemm_wmma<128, true><<<gblocks, 128, 0, stream>>>(agg, norm_d, W2, b2, h, n);
  zero_f32<<<1, 256, 0, stream>>>(stats, 256);
  bn_stats<<<1024, 128, 0, stream>>>(h, stats, n);
  bn_scale<<<4096, 128, 0, stream>>>(h, stats, g2, be2, norm_s, hs, n);

  // ---- layer 3: conv -> log_softmax ----
  zero_f32<<<4096, 256, 0, stream>>>(agg, (size_t)n * 128);
  spmm_kernel<<<2048, 256, 0, stream>>>(hs, src, dst, agg, ne);
  gemm_wmma<64, false><<<gblocks, 128, 0, stream>>>(agg, norm_d, W3, b3, h, n);
  logsoftmax_kernel<<<2048, 256, 0, stream>>>(h, out, n);
}